// BoundaryAwareDiceLoss_65884798321056
// MI455X (gfx1250) — compile-verified
//
#include <hip/hip_runtime.h>
#include <hip/hip_bf16.h>

// Problem geometry (fixed by the reference): [B,C,H,W] = [32,1,512,512]
#define W_DIM   512
#define H_DIM   512
#define B_DIM   32
#define N_ELEM  (B_DIM * H_DIM * W_DIM)          // 8,388,608
#define NG      (N_ELEM / 4)                     // float4 groups: 2,097,152
#define GPR     (W_DIM / 4)                      // groups per row: 128
#define GPI     (H_DIM * GPR)                    // groups per image: 65,536
#define NBLK    2048
#define NTHR    256
#define NITER   (NG / (NBLK * NTHR))             // exactly 4 groups per thread

typedef __attribute__((ext_vector_type(2))) float v2f;
typedef __attribute__((ext_vector_type(8))) float v8f;
typedef __attribute__((ext_vector_type(4))) int   v4i;

typedef __attribute__((address_space(1))) v4i* gv4i_p;   // global int4*
typedef __attribute__((address_space(3))) v4i* lv4i_p;   // LDS int4*

__device__ __forceinline__ float wave_reduce(float v) {
    v += __shfl_down(v, 16, 32);
    v += __shfl_down(v, 8, 32);
    v += __shfl_down(v, 4, 32);
    v += __shfl_down(v, 2, 32);
    v += __shfl_down(v, 1, 32);
    return v;
}

__device__ __forceinline__ void unpack4(const float4& v, float* a) {
    a[0] = v.x; a[1] = v.y; a[2] = v.z; a[3] = v.w;
}

// Kernel 1: streaming partial reduction. Each block writes 5 partial sums:
// [0]=sum(p*t) [1]=sum(p) [2]=sum(t) [3]=sum(bce_i) [4]=sum(boundary)
__global__ void __launch_bounds__(NTHR)
bdl_partial_kernel(const float* __restrict__ pred,
                   const float* __restrict__ targ,
                   float* __restrict__ ws) {
    const float4* __restrict__ p4 = (const float4*)pred;
    const float4* __restrict__ t4 = (const float4*)targ;

    float s_pt = 0.f, s_p = 0.f, s_t = 0.f, s_bce = 0.f, s_bnd = 0.f;

    const int g0 = blockIdx.x * NTHR + threadIdx.x;

#pragma unroll
    for (int k = 0; k < NITER; ++k) {
        const int g  = g0 + k * (NBLK * NTHR);   // always < NG (exact tiling)
        const int wg = g & (GPR - 1);
        const int r  = g >> 7;                   // row index (b*H + h)
        const int h  = r & (H_DIM - 1);
        const int b  = r >> 9;

        const float4 tc  = t4[g];
        const float4 pc  = p4[g];
        // clamped-shift semantics: at an edge the "neighbor" is self,
        // which contributes nothing (fg & !fg == 0) -> offset 0 is correct.
        const float4 tu  = t4[g + ((h == 0)         ? 0 : -GPR)];
        const float4 td  = t4[g + ((h == H_DIM - 1) ? 0 :  GPR)];
        const float4 tb0 = t4[g + ((b == 0)         ? 0 : -GPI)];
        const float4 tb1 = t4[g + ((b == B_DIM - 1) ? 0 :  GPI)];

        const float left  = (wg == 0)       ? tc.x : targ[(g << 2) - 1];
        const float right = (wg == GPR - 1) ? tc.w : targ[(g << 2) + 4];

        float tcv[4], pcv[4], tuv[4], tdv[4], b0v[4], b1v[4];
        unpack4(tc, tcv); unpack4(pc, pcv); unpack4(tu, tuv);
        unpack4(td, tdv); unpack4(tb0, b0v); unpack4(tb1, b1v);
        const float lv[4] = { left, tcv[0], tcv[1], tcv[2] };
        const float rv[4] = { tcv[1], tcv[2], tcv[3], right };

#pragma unroll
        for (int e = 0; e < 4; ++e) {
            const float t = tcv[e];
            const float p = pcv[e];
            s_pt += p * t;
            s_p  += p;
            s_t  += t;

            const bool fg = t > 0.5f;
            const bool bg = (lv[e]  <= 0.5f) | (rv[e]  <= 0.5f) |
                            (tuv[e] <= 0.5f) | (tdv[e] <= 0.5f) |
                            (b0v[e] <= 0.5f) | (b1v[e] <= 0.5f);
            s_bnd += (fg & bg) ? 1.0f : 0.0f;

            const float pp = fminf(fmaxf(p, 1e-7f), 1.0f - 1e-7f);
            s_bce -= t * logf(pp) + (1.0f - t) * log1pf(-pp);
        }
    }

    __shared__ float red[5][NTHR / 32];
    const int lane = threadIdx.x & 31;
    const int wid  = threadIdx.x >> 5;

    const float v0 = wave_reduce(s_pt);
    const float v1 = wave_reduce(s_p);
    const float v2 = wave_reduce(s_t);
    const float v3 = wave_reduce(s_bce);
    const float v4 = wave_reduce(s_bnd);
    if (lane == 0) {
        red[0][wid] = v0; red[1][wid] = v1; red[2][wid] = v2;
        red[3][wid] = v3; red[4][wid] = v4;
    }
    __syncthreads();
    if (threadIdx.x < 5) {
        float s = 0.f;
#pragma unroll
        for (int w = 0; w < NTHR / 32; ++w) s += red[threadIdx.x][w];
        ws[threadIdx.x * NBLK + blockIdx.x] = s;
    }
}

// Kernel 2: async-copy the 5*2048 partials into LDS, then reduce each
// 2048-vector with v_wmma_f32_16x16x4_f32 (A = data tile, B = ones =>
// row-sum accumulation in C), finalize the scalar on thread 0.
__global__ void __launch_bounds__(NTHR)
bdl_finalize_kernel(const float* __restrict__ ws, float* __restrict__ out) {
    __shared__ float smem[5 * NBLK];
    __shared__ float res[5];

    // 5*2048 floats = 2560 float4 chunks, async global->LDS
    float* wsnc = const_cast<float*>(ws);
    for (int i = threadIdx.x; i < (5 * NBLK) / 4; i += NTHR) {
        __builtin_amdgcn_global_load_async_to_lds_b128(
            (gv4i_p)(wsnc + i * 4), (lv4i_p)(&smem[i * 4]), 0, 0);
    }
#if __has_builtin(__builtin_amdgcn_s_wait_asynccnt)
    __builtin_amdgcn_s_wait_asynccnt(0);
#else
    asm volatile("s_wait_asynccnt 0" ::: "memory");
#endif
    __syncthreads();

    const int lane = threadIdx.x & 31;
    const int q    = threadIdx.x >> 5;   // wave id; waves 0..4 each own one sum

    if (q < 5) {
        // A is 16x4 f32: VGPR0 = {A[m,0] lanes0-15, A[m,2] lanes16-31},
        // VGPR1 = {A[m,1], A[m,3]}. With A[m,k] = v[4m+k]:
        // lane L<16 holds v[4L], v[4L+1]; lane L>=16 holds v[4(L-16)+2], v[4(L-16)+3].
        const int lsel = (lane < 16) ? (4 * lane) : (4 * (lane - 16) + 2);
        const int base = q * NBLK;

        v8f acc = {0.f, 0.f, 0.f, 0.f, 0.f, 0.f, 0.f, 0.f};
        v2f bones; bones.x = 1.0f; bones.y = 1.0f;   // B = all-ones 4x16

        for (int c = 0; c < NBLK / 64; ++c) {        // 32 chunks of 64 values
            const float2 av = *(const float2*)(&smem[base + c * 64 + lsel]);
            v2f a; a.x = av.x; a.y = av.y;
            // D[m,n] = sum_k A[m,k]*1 + C[m,n]  (row-sum accumulate)
            acc = __builtin_amdgcn_wmma_f32_16x16x4_f32(
                false, a, false, bones, (short)0, acc, false, false);
        }
        // D layout: VGPR r holds D[m=r, n=lane] (lanes0-15) / D[m=8+r, n=lane-16].
        // Sum of all D over m for one column = per-lane vsum + (lane+16)'s vsum.
        float s = acc[0] + acc[1] + acc[2] + acc[3] +
                  acc[4] + acc[5] + acc[6] + acc[7];
        s += __shfl_down(s, 16, 32);
        if (lane == 0) res[q] = s;
    }
    __syncthreads();

    if (threadIdx.x == 0) {
        const float s_pt  = res[0];
        const float s_p   = res[1];
        const float s_t   = res[2];
        const float s_bce = res[3];
        const float s_bnd = res[4];
        const float smooth = 1e-5f;
        const float dice = 1.0f - (2.0f * s_pt + smooth) / (s_p + s_t + smooth);
        const float invN = 1.0f / (float)N_ELEM;
        out[0] = dice + 10.0f * (s_bce * invN) * (s_bnd * invN);
    }
}

extern "C" void kernel_launch(void* const* d_in, const int* in_sizes, int n_in,
                              void* d_out, int out_size, void* d_ws, size_t ws_size,
                              hipStream_t stream) {
    (void)in_sizes; (void)n_in; (void)out_size; (void)ws_size;
    const float* pred = (const float*)d_in[0];
    const float* targ = (const float*)d_in[1];
    float* ws  = (float*)d_ws;    // 5 * NBLK floats = 40 KB, fully overwritten
    float* out = (float*)d_out;

    bdl_partial_kernel<<<NBLK, NTHR, 0, stream>>>(pred, targ, ws);
    bdl_finalize_kernel<<<1, NTHR, 0, stream>>>(ws, out);
}